// MinPlus_70832600646269
// MI455X (gfx1250) — compile-verified
//
#include <hip/hip_runtime.h>

// Tropical (min,+) matmul: out[b,j] = min_k ( X[b,k] + W[j,k] )
// B=1024, K=512, J=512, fp32.
//
// VALU-bound problem (537M add+min ops vs only ~5MB of HBM traffic), so:
//  - 64x64 output tile per 256-thread block, 4x4 register micro-tile/thread
//  - k processed in pairs: float2 adds (-> v_pk_add_f32) + fminf(fminf(..))
//    (-> v_min3_num_f32): ~1 VALU instr per (add,min) pair
//  - CDNA5 async global->LDS loads (ASYNCcnt) with double buffering;
//    W tile is transposed on the fly via per-lane b32 async loads so that
//    compute-side LDS reads are contiguous float4 (conflict-free).

#define B_DIM 1024
#define K_DIM 512
#define J_DIM 512
#define BT 64   // b-tile per block
#define JT 64   // j-tile per block
#define KT 32   // k-chunk staged in LDS
#define NTHREADS 256

// Low 32 bits of a generic LDS pointer are the workgroup-relative LDS offset
__device__ __forceinline__ unsigned lds_off(const void* p) {
    return (unsigned)(unsigned long long)(uintptr_t)p;
}

// CDNA5 async tensor-path loads: memory -> LDS, tracked with ASYNCcnt.
__device__ __forceinline__ void async_b128(unsigned loff, const float* g) {
    unsigned long long ga = (unsigned long long)(uintptr_t)g;
    asm volatile("global_load_async_to_lds_b128 %0, %1, off"
                 :: "v"(loff), "v"(ga) : "memory");
}
__device__ __forceinline__ void async_b32(unsigned loff, const float* g) {
    unsigned long long ga = (unsigned long long)(uintptr_t)g;
    asm volatile("global_load_async_to_lds_b32 %0, %1, off"
                 :: "v"(loff), "v"(ga) : "memory");
}
__device__ __forceinline__ void wait_async0() {
    asm volatile("s_wait_asynccnt 0" ::: "memory");
}

__global__ __launch_bounds__(NTHREADS)
void MinPlus_70832600646269_kernel(const float* __restrict__ X,
                                   const float* __restrict__ W,
                                   float* __restrict__ out) {
    // Double-buffered tiles: X row-major [b][k], W transposed [k][j].
    __shared__ float Xs[2][BT * KT];   // 8 KB each
    __shared__ float Ws[2][KT * JT];   // 8 KB each  -> 32 KB total

    const int tid = threadIdx.x;
    const int tx  = tid & 15;          // j micro-tile index (0..15)
    const int ty  = tid >> 4;          // b micro-tile index (0..15)
    const int b0  = blockIdx.y * BT;
    const int j0  = blockIdx.x * JT;

    auto load_tiles = [&](int buf, int k0) {
        // X tile: 64 rows x 32 floats = 512 x 16B chunks; 2 b128 per thread.
        #pragma unroll
        for (int i = 0; i < 2; ++i) {
            int c   = tid + i * NTHREADS;   // 0..511
            int row = c >> 3;               // 8 chunks per 32-float row
            int col = (c & 7) << 2;         // float offset, 16B aligned
            async_b128(lds_off(&Xs[buf][row * KT + col]),
                       X + (b0 + row) * K_DIM + (k0 + col));
        }
        // W tile transposed: Ws[k][j] = W[j0+j][k0+k]; 8 b32 per thread.
        // Lanes walk k (contiguous global reads), writes scatter into [k][j].
        int lk = tid & 31;                  // k offset within chunk
        int jg = tid >> 5;                  // 0..7
        #pragma unroll
        for (int i = 0; i < 8; ++i) {
            int j = jg * 8 + i;             // 0..63
            async_b32(lds_off(&Ws[buf][lk * JT + j]),
                      W + (j0 + j) * K_DIM + (k0 + lk));
        }
    };

    float acc[4][4];
    #pragma unroll
    for (int r = 0; r < 4; ++r)
        #pragma unroll
        for (int c = 0; c < 4; ++c)
            acc[r][c] = __builtin_inff();

    load_tiles(0, 0);
    wait_async0();
    __syncthreads();

    int buf = 0;
    const int NK = K_DIM / KT;
    for (int kc = 0; kc < NK; ++kc) {
        if (kc + 1 < NK) load_tiles(buf ^ 1, (kc + 1) * KT);

        const float* xs = &Xs[buf][0];
        const float* ws = &Ws[buf][0];
        #pragma unroll
        for (int kk = 0; kk < KT; kk += 2) {
            float2 xv[4];
            #pragma unroll
            for (int r = 0; r < 4; ++r)   // broadcast reads, conflict-free
                xv[r] = *reinterpret_cast<const float2*>(
                            &xs[(ty * 4 + r) * KT + kk]);
            // contiguous float4 per lane, conflict-free
            float4 w0 = *reinterpret_cast<const float4*>(&ws[kk * JT + tx * 4]);
            float4 w1 = *reinterpret_cast<const float4*>(&ws[(kk + 1) * JT + tx * 4]);
            float2 wp[4] = { {w0.x, w1.x}, {w0.y, w1.y},
                             {w0.z, w1.z}, {w0.w, w1.w} };
            #pragma unroll
            for (int r = 0; r < 4; ++r) {
                #pragma unroll
                for (int c = 0; c < 4; ++c) {
                    // float2 add -> v_pk_add_f32; min-of-three -> v_min3_num_f32
                    float t0 = xv[r].x + wp[c].x;
                    float t1 = xv[r].y + wp[c].y;
                    acc[r][c] = fminf(acc[r][c], fminf(t0, t1));
                }
            }
        }

        wait_async0();       // async LDS writes complete for next buffer
        __syncthreads();     // whole WG done reading current buffer
        buf ^= 1;
    }

    #pragma unroll
    for (int r = 0; r < 4; ++r) {
        float4 o = make_float4(acc[r][0], acc[r][1], acc[r][2], acc[r][3]);
        *reinterpret_cast<float4*>(
            &out[(b0 + ty * 4 + r) * J_DIM + (j0 + tx * 4)]) = o;
    }
}

extern "C" void kernel_launch(void* const* d_in, const int* in_sizes, int n_in,
                              void* d_out, int out_size, void* d_ws, size_t ws_size,
                              hipStream_t stream) {
    const float* X = (const float*)d_in[0];   // [1024][512]
    const float* W = (const float*)d_in[1];   // [512][512]
    float* out = (float*)d_out;               // [1024][512]
    dim3 grid(J_DIM / JT, B_DIM / BT);        // (8, 16) = 128 blocks
    MinPlus_70832600646269_kernel<<<grid, NTHREADS, 0, stream>>>(X, W, out);
}